// Projection_73169062855068
// MI455X (gfx1250) — compile-verified
//
#include <hip/hip_runtime.h>
#include <hip/hip_bf16.h>

#define RDIM 128
#define CIN 64
#define COUT 64
#define KIN 69          // CIN + 5
#define KPAD 72         // padded to multiple of 4 for 16x16x4 wmma
#define FSTRIDE 76      // LDS feat row stride (floats), conflict-free
#define BN_EPS 1e-5f

typedef __attribute__((ext_vector_type(2))) float v2f;
typedef __attribute__((ext_vector_type(8))) float v8f;

// ---------------------------------------------------------------------------
// Kernel 0: zero workspace accumulators + output (output must start at 0.0f
// for the uint-bitpattern atomicMax segment-max).
// ---------------------------------------------------------------------------
__global__ void k_zero(float* __restrict__ ws, int nws,
                       float* __restrict__ out, int nout) {
    int stride = gridDim.x * blockDim.x;
    int i = blockIdx.x * blockDim.x + threadIdx.x;
    for (int j = i; j < nws;  j += stride) ws[j]  = 0.0f;
    for (int j = i; j < nout; j += stride) out[j] = 0.0f;
}

// ---------------------------------------------------------------------------
// Kernel 1: per-point scatter of norm_coords into pillar accumulators
// pillarAcc[seg] = {sum_x, sum_y, sum_z, count}
// ---------------------------------------------------------------------------
__global__ void k_pillar_sum(const float* __restrict__ nc,
                             const int*   __restrict__ ci,
                             const int*   __restrict__ paPtr,
                             float* __restrict__ pillarAcc, int N) {
    int i = blockIdx.x * blockDim.x + threadIdx.x;
    if (i >= N) return;
    int pa = *paPtr;
    int ax[3]; { int k = 0; for (int a = 0; a < 4; ++a) if (a != pa) ax[k++] = a; }
    int c0 = ci[i * 4 + ax[0]];
    int c1 = ci[i * 4 + ax[1]];
    int c2 = ci[i * 4 + ax[2]];
    int seg = (c0 * RDIM + c1) * RDIM + c2;
    atomicAdd(&pillarAcc[seg * 4 + 0], nc[i * 3 + 0]);
    atomicAdd(&pillarAcc[seg * 4 + 1], nc[i * 3 + 1]);
    atomicAdd(&pillarAcc[seg * 4 + 2], nc[i * 3 + 2]);
    atomicAdd(&pillarAcc[seg * 4 + 3], 1.0f);
}

// ---------------------------------------------------------------------------
// Kernel 2 (run twice): fused feature-assembly + WMMA GEMM.
//   pass 1: accumulate per-channel sum / sumsq of h = feat@W.T + b
//   pass 2: h -> scale*acc + shift (bias & BN folded), ReLU, atomicMax scatter
// One wave owns one 16-row tile; 8 waves / block.
//
// Staged-weight layout (pair-interleaved so each B fragment is ONE aligned
// ds_load_b64):  sW[((k>>2)*64 + n)*4 + (k&3)] = W^T[k][n]  (zero-padded).
// A lane (half h, col n) needs K = 4*kk + 2h, 4*kk + 2h + 1  ->  contiguous.
// ---------------------------------------------------------------------------
__global__ __launch_bounds__(256)
void k_gemm(const float* __restrict__ features,
            const float* __restrict__ nc,
            const int*   __restrict__ ci,
            const float* __restrict__ pv,
            const float* __restrict__ W,
            const float* __restrict__ bias,
            const int*   __restrict__ paPtr,
            const float* __restrict__ pillarAcc,
            float* __restrict__ bnSum,       // [64]  (pass 1)
            float* __restrict__ bnSq,        // [64]  (pass 1)
            const float* __restrict__ scale, // [64]  (pass 2)
            const float* __restrict__ shift, // [64]  (pass 2)
            float* __restrict__ out,         // (pass 2) B*R*R*64, >=0 via bits
            int N, int Np, int numTiles, int pass) {
    __shared__ float sW[KPAD * COUT];        // pair-interleaved W^T (18 KB)
    __shared__ float sF[8][16 * FSTRIDE];    // per-wave 16x72 feat tile
    __shared__ int   sSeg[8][16];
    __shared__ float sBn[2 * COUT];

    const int tid  = threadIdx.x;
    const int wave = tid >> 5;
    const int lane = tid & 31;
    const int half = lane >> 4;      // 0: lanes 0-15, 1: lanes 16-31
    const int l16  = lane & 15;

    // Stage W^T (padded) into LDS in pair-interleaved layout.
    // dst = (kk*64 + n)*4 + kl  with k = kk*4 + kl
    for (int dst = tid; dst < KPAD * COUT; dst += 256) {
        int kl = dst & 3;
        int n  = (dst >> 2) & 63;
        int kk = dst >> 8;
        int k  = kk * 4 + kl;
        sW[dst] = (k < KIN) ? W[n * KIN + k] : 0.0f;
    }
    if (tid < 2 * COUT) sBn[tid] = 0.0f;

    int pa = *paPtr;
    int ax[3]; { int k = 0; for (int a = 0; a < 4; ++a) if (a != pa) ax[k++] = a; }

    const int tile = blockIdx.x * 8 + wave;
    const bool active = (tile < numTiles);
    float* fw = &sF[wave][0];

    if (active) {
        int row = tile * 16 + l16;
        if (row >= N) row = N - 1;          // N is a multiple of 16 here
        int b  = row / Np;
        int pp = row - b * Np;
        // 64 channels: coalesced gather (16 consecutive pp per half-wave)
        const float* fbase = features + (size_t)b * CIN * Np + pp;
        for (int c = half; c < CIN; c += 2)
            fw[l16 * FSTRIDE + c] = fbase[(size_t)c * Np];
        if (half == 0) {
            float xp = pv[row * 4 + ax[1]];
            float yp = pv[row * 4 + ax[2]];
            int c0 = ci[row * 4 + ax[0]];
            int c1 = ci[row * 4 + ax[1]];
            int c2 = ci[row * 4 + ax[2]];
            int seg = (c0 * RDIM + c1) * RDIM + c2;
            sSeg[wave][l16] = seg;
            float sx = pillarAcc[seg * 4 + 0];
            float sy = pillarAcc[seg * 4 + 1];
            float sz = pillarAcc[seg * 4 + 2];
            float inv = 1.0f / fmaxf(pillarAcc[seg * 4 + 3], 1.0f);
            float* fr = fw + l16 * FSTRIDE;
            fr[64] = xp;
            fr[65] = yp;
            fr[66] = nc[row * 3 + 0] - sx * inv;
            fr[67] = nc[row * 3 + 1] - sy * inv;
            fr[68] = nc[row * 3 + 2] - sz * inv;
            fr[69] = 0.0f; fr[70] = 0.0f; fr[71] = 0.0f;
        }
    }
    __syncthreads();

    // ---- WMMA: 18 k-steps of 16x16x4 f32, 4 N-tiles of 16 channels ----
    v8f acc[4] = {};
    if (active) {
        const int koff = half * 2;   // A: lanes<16 hold K,K+1; >=16 hold K+2,K+3
        const float* frow = fw + l16 * FSTRIDE + koff;     // 8B aligned
        const float* wrow = sW + l16 * 4 + koff;           // 8B aligned
        #pragma unroll
        for (int kk = 0; kk < KPAD / 4; ++kk) {
            v2f a  = *(const v2f*)(frow + kk * 4);
            const float* wb = wrow + kk * (COUT * 4);
            v2f b0 = *(const v2f*)(wb);
            v2f b1 = *(const v2f*)(wb + 64);
            v2f b2 = *(const v2f*)(wb + 128);
            v2f b3 = *(const v2f*)(wb + 192);
            acc[0] = __builtin_amdgcn_wmma_f32_16x16x4_f32(false, a, false, b0, (short)0, acc[0], false, false);
            acc[1] = __builtin_amdgcn_wmma_f32_16x16x4_f32(false, a, false, b1, (short)0, acc[1], false, false);
            acc[2] = __builtin_amdgcn_wmma_f32_16x16x4_f32(false, a, false, b2, (short)0, acc[2], false, false);
            acc[3] = __builtin_amdgcn_wmma_f32_16x16x4_f32(false, a, false, b3, (short)0, acc[3], false, false);
        }
    }

    if (pass == 1) {
        // BN statistics over h = acc + bias. C layout: lane<16 -> rows j,
        // lane>=16 -> rows 8+j; both halves share channel l16 within a tile.
        if (active) {
            #pragma unroll
            for (int t = 0; t < 4; ++t) {
                int ch = t * 16 + l16;
                float bch = bias[ch];
                float s = 0.0f, q = 0.0f;
                #pragma unroll
                for (int j = 0; j < 8; ++j) {
                    float h = acc[t][j] + bch;
                    s += h; q += h * h;
                }
                s += __shfl_xor(s, 16, 32);
                q += __shfl_xor(q, 16, 32);
                if (half == 0) {
                    atomicAdd(&sBn[ch], s);
                    atomicAdd(&sBn[COUT + ch], q);
                }
            }
        }
        __syncthreads();
        if (tid < COUT) {
            atomicAdd(&bnSum[tid], sBn[tid]);
            atomicAdd(&bnSq[tid],  sBn[COUT + tid]);
        }
    } else {
        // BN + ReLU + segment_max via uint atomicMax (values >= 0)
        if (active) {
            #pragma unroll
            for (int t = 0; t < 4; ++t) {
                int ch = t * 16 + l16;
                float sc = scale[ch];
                float sh = shift[ch];
                #pragma unroll
                for (int j = 0; j < 8; ++j) {
                    int rowm = half * 8 + j;
                    int seg  = sSeg[wave][rowm];
                    float h  = fmaxf(acc[t][j] * sc + sh, 0.0f);
                    atomicMax((unsigned int*)&out[(size_t)seg * COUT + ch],
                              __float_as_uint(h));
                }
            }
        }
    }
}

// ---------------------------------------------------------------------------
// Kernel 3: finalize BN -> per-channel scale/shift (bias folded into shift)
// ---------------------------------------------------------------------------
__global__ void k_bnfin(const float* __restrict__ bnSum,
                        const float* __restrict__ bnSq,
                        const float* __restrict__ gamma,
                        const float* __restrict__ beta,
                        const float* __restrict__ bias,
                        float* __restrict__ scale,
                        float* __restrict__ shift, float invN) {
    int c = threadIdx.x;
    if (c < COUT) {
        float mu  = bnSum[c] * invN;
        float var = bnSq[c] * invN - mu * mu;
        float sc  = gamma[c] * rsqrtf(var + BN_EPS);
        scale[c] = sc;
        shift[c] = (bias[c] - mu) * sc + beta[c];
    }
}

// ---------------------------------------------------------------------------
extern "C" void kernel_launch(void* const* d_in, const int* in_sizes, int n_in,
                              void* d_out, int out_size, void* d_ws, size_t ws_size,
                              hipStream_t stream) {
    const float* features = (const float*)d_in[0];
    const float* nc       = (const float*)d_in[1];
    const int*   ci       = (const int*)  d_in[2];
    const float* pv       = (const float*)d_in[3];
    const float* W        = (const float*)d_in[4];
    const float* bias     = (const float*)d_in[5];
    const float* gamma    = (const float*)d_in[6];
    const float* beta     = (const float*)d_in[7];
    const int*   paPtr    = (const int*)  d_in[8];
    float* out = (float*)d_out;

    const int N  = in_sizes[1] / 3;                    // 400000
    const int B  = out_size / (RDIM * RDIM * COUT);    // 2
    const int Np = N / B;

    // workspace layout (floats)
    float* ws       = (float*)d_ws;
    const int nseg  = B * RDIM * RDIM;                 // 32768
    float* pillar   = ws;                              // nseg*4
    float* bnSum    = ws + nseg * 4;                   // 64
    float* bnSq     = bnSum + COUT;                    // 64
    float* scale    = bnSq + COUT;                     // 64
    float* shift    = scale + COUT;                    // 64
    const int nZero = nseg * 4 + 2 * COUT;             // accumulators only

    k_zero<<<2048, 256, 0, stream>>>(ws, nZero, out, out_size);

    k_pillar_sum<<<(N + 255) / 256, 256, 0, stream>>>(nc, ci, paPtr, pillar, N);

    const int numTiles = (N + 15) / 16;
    const int nBlk     = (numTiles + 7) / 8;

    k_gemm<<<nBlk, 256, 0, stream>>>(features, nc, ci, pv, W, bias, paPtr,
                                     pillar, bnSum, bnSq, scale, shift, out,
                                     N, Np, numTiles, /*pass=*/1);

    k_bnfin<<<1, 64, 0, stream>>>(bnSum, bnSq, gamma, beta, bias, scale, shift,
                                  1.0f / (float)N);

    k_gemm<<<nBlk, 256, 0, stream>>>(features, nc, ci, pv, W, bias, paPtr,
                                     pillar, bnSum, bnSq, scale, shift, out,
                                     N, Np, numTiles, /*pass=*/2);
}